// IGMC_23605140258904
// MI455X (gfx1250) — compile-verified
//
#include <hip/hip_runtime.h>
#include <math.h>

#define N_NODES 100000
#define N_EDGES 3200000
#define R_REL   5
#define B_BASES 2
#define IN_FEAT 4
#define HID     32
#define G_PAIRS 25000
#define MLP_HID 128
#define CAT     (4 * HID)   // 128

typedef float v2f __attribute__((ext_vector_type(2)));
typedef float v8f __attribute__((ext_vector_type(8)));

__device__ __forceinline__ void atomic_add_f32_nr(float* p, float v) {
    // relaxed, device scope -> non-returning global_atomic_add_f32
    __hip_atomic_fetch_add(p, v, __ATOMIC_RELAXED, __HIP_MEMORY_SCOPE_AGENT);
}

// ---------------------------------------------------------------------------
// W[r] = sum_b comp[r,b] * bases[b]   (slot R = self-loop weight)
// ---------------------------------------------------------------------------
__global__ void build_W(const float* __restrict__ bases,
                        const float* __restrict__ comp,
                        const float* __restrict__ loopw,
                        float* __restrict__ Wall,
                        int in_dim) {
    const int per   = in_dim * HID;
    const int total = (R_REL + 1) * per;
    int idx = blockIdx.x * blockDim.x + threadIdx.x;
    if (idx >= total) return;
    int r  = idx / per;
    int io = idx % per;
    if (r < R_REL) {
        float acc = 0.f;
#pragma unroll
        for (int b = 0; b < B_BASES; ++b)
            acc += comp[r * B_BASES + b] * bases[b * per + io];
        Wall[idx] = acc;
    } else {
        Wall[idx] = loopw[io];
    }
}

// ---------------------------------------------------------------------------
// xr[r] = X @ W[r]  for r in [0,R); selfbuf = X @ loopw  (rel == R)
// One wave (32 lanes) computes one 16x16 f32 tile via v_wmma_f32_16x16x4_f32.
// A layout (16x4 f32): lanes 0-15 -> M=lane, VGPR0=K0,VGPR1=K1;
//                      lanes 16-31 -> M=lane-16, VGPR0=K2,VGPR1=K3.
// C/D layout: VGPR v: lanes 0-15 -> M=v, lanes 16-31 -> M=v+8; N=lane%16.
// ---------------------------------------------------------------------------
__global__ void node_gemm_wmma(const float* __restrict__ X, int x_stride, int in_dim,
                               const float* __restrict__ Wall,
                               float* __restrict__ xr,
                               float* __restrict__ selfbuf) {
    const int lane = threadIdx.x & 31;
    const int wave = (blockIdx.x * blockDim.x + threadIdx.x) >> 5;
    const int rowTiles   = (N_NODES + 15) / 16;
    const int outTiles   = HID / 16;                    // 2
    const int totalTiles = rowTiles * outTiles * (R_REL + 1);
    if (wave >= totalTiles) return;

    int t = wave;
    const int rel = t % (R_REL + 1); t /= (R_REL + 1);
    const int ct  = t % outTiles;    t /= outTiles;
    const int rt  = t;

    const int row0 = rt * 16;
    const int col0 = ct * 16;
    const int half = lane >> 4;
    const int m    = lane & 15;
    const int row  = row0 + m;
    const bool rowOK = (row < N_NODES);

    const float* W = Wall + (size_t)rel * in_dim * HID;   // [in_dim, HID] row-major

    v8f acc = {};
    for (int k0 = 0; k0 < in_dim; k0 += 4) {
        const int ka = k0 + 2 * half;
        v2f a, b;
        a.x = rowOK ? X[(size_t)row * x_stride + ka]     : 0.f;
        a.y = rowOK ? X[(size_t)row * x_stride + ka + 1] : 0.f;
        b.x = W[(size_t)ka       * HID + col0 + m];
        b.y = W[(size_t)(ka + 1) * HID + col0 + m];
        acc = __builtin_amdgcn_wmma_f32_16x16x4_f32(false, a, false, b,
                                                    (short)0, acc, false, false);
    }

    float* dst = (rel < R_REL) ? (xr + (size_t)rel * N_NODES * HID) : selfbuf;
#pragma unroll
    for (int v = 0; v < 8; ++v) {
        const int orow = row0 + v + 8 * half;
        if (orow < N_NODES)
            dst[(size_t)orow * HID + col0 + m] = acc[v];
    }
}

// ---------------------------------------------------------------------------
// Wave per edge, lane per feature: agg[dst] += xr[etype, src]   (f32 atomics)
// ---------------------------------------------------------------------------
__global__ void edge_scatter(const int* __restrict__ src,
                             const int* __restrict__ dst,
                             const int* __restrict__ etype,
                             const float* __restrict__ xr,
                             float* __restrict__ agg) {
    const long long tid = (long long)blockIdx.x * blockDim.x + threadIdx.x;
    const int e    = (int)(tid >> 5);
    const int lane = (int)(tid & 31);
    if (e >= N_EDGES) return;
    const int s = src[e];
    const int d = dst[e];
    const int r = etype[e];
    const float v = xr[((size_t)r * N_NODES + s) * HID + lane];
    atomic_add_f32_nr(agg + (size_t)d * HID + lane, v);
}

// ---------------------------------------------------------------------------
// h = tanh(agg + self + bias); write into concat[:, l*32 : l*32+32]
// ---------------------------------------------------------------------------
__global__ void finalize_layer(const float* __restrict__ agg,
                               const float* __restrict__ selfbuf,
                               const float* __restrict__ bias,
                               float* __restrict__ concat, int layer) {
    int idx = blockIdx.x * blockDim.x + threadIdx.x;
    if (idx >= N_NODES * HID) return;
    const int n = idx / HID;
    const int c = idx % HID;
    const float h = tanhf(agg[idx] + selfbuf[idx] + bias[c]);
    concat[(size_t)n * CAT + layer * HID + c] = h;
}

// ---------------------------------------------------------------------------
// hidden = relu([concat[user] | concat[item]] @ w1 + bl1)  via WMMA, K=256
// ---------------------------------------------------------------------------
__global__ void mlp1_wmma(const float* __restrict__ concat,
                          const int* __restrict__ user_ids,
                          const int* __restrict__ item_ids,
                          const float* __restrict__ w1,
                          const float* __restrict__ bl1,
                          float* __restrict__ hidden) {
    const int lane = threadIdx.x & 31;
    const int wave = (blockIdx.x * blockDim.x + threadIdx.x) >> 5;
    const int rowTiles   = (G_PAIRS + 15) / 16;
    const int colTiles   = MLP_HID / 16;                 // 8
    const int totalTiles = rowTiles * colTiles;
    if (wave >= totalTiles) return;

    const int ct = wave % colTiles;
    const int rt = wave / colTiles;
    const int row0 = rt * 16;
    const int col0 = ct * 16;
    const int half = lane >> 4;
    const int m    = lane & 15;
    const int row  = row0 + m;
    const bool rowOK = (row < G_PAIRS);

    const int uid = rowOK ? user_ids[row] : 0;
    const int iid = rowOK ? item_ids[row] : 0;
    const float* urow = concat + (size_t)uid * CAT;
    const float* irow = concat + (size_t)iid * CAT;

    v8f acc = {};
    for (int k0 = 0; k0 < 2 * CAT; k0 += 4) {
        const int ka = k0 + 2 * half;
        const float a0 = (ka     < CAT) ? urow[ka]           : irow[ka - CAT];
        const float a1 = (ka + 1 < CAT) ? urow[ka + 1]       : irow[ka + 1 - CAT];
        v2f a, b;
        a.x = rowOK ? a0 : 0.f;
        a.y = rowOK ? a1 : 0.f;
        b.x = w1[(size_t)ka       * MLP_HID + col0 + m];
        b.y = w1[(size_t)(ka + 1) * MLP_HID + col0 + m];
        acc = __builtin_amdgcn_wmma_f32_16x16x4_f32(false, a, false, b,
                                                    (short)0, acc, false, false);
    }

    const float bcol = bl1[col0 + m];
#pragma unroll
    for (int v = 0; v < 8; ++v) {
        const int orow = row0 + v + 8 * half;
        if (orow < G_PAIRS) {
            const float h = acc[v] + bcol;
            hidden[(size_t)orow * MLP_HID + col0 + m] = h > 0.f ? h : 0.f;
        }
    }
}

// ---------------------------------------------------------------------------
// out[g] = dot(hidden[g], w2) + bl2   (wave per row, wave32 shfl reduction)
// ---------------------------------------------------------------------------
__global__ void mlp2_dot(const float* __restrict__ hidden,
                         const float* __restrict__ w2,
                         const float* __restrict__ bl2,
                         float* __restrict__ out) {
    const long long tid = (long long)blockIdx.x * blockDim.x + threadIdx.x;
    const int g    = (int)(tid >> 5);
    const int lane = (int)(tid & 31);
    if (g >= G_PAIRS) return;
    float s = 0.f;
#pragma unroll
    for (int k = lane; k < MLP_HID; k += 32)
        s += hidden[(size_t)g * MLP_HID + k] * w2[k];
#pragma unroll
    for (int off = 16; off > 0; off >>= 1)
        s += __shfl_xor(s, off, 32);
    if (lane == 0) out[g] = s + bl2[0];
}

// ---------------------------------------------------------------------------
extern "C" void kernel_launch(void* const* d_in, const int* in_sizes, int n_in,
                              void* d_out, int out_size, void* d_ws, size_t ws_size,
                              hipStream_t stream) {
    (void)in_sizes; (void)n_in; (void)out_size; (void)ws_size;

    const float* x        = (const float*)d_in[0];
    const int*   src      = (const int*)d_in[1];
    const int*   dst      = (const int*)d_in[2];
    const int*   etype    = (const int*)d_in[3];
    const int*   user_ids = (const int*)d_in[4];
    const int*   item_ids = (const int*)d_in[5];
    const float* w1       = (const float*)d_in[22];
    const float* bl1      = (const float*)d_in[23];
    const float* w2       = (const float*)d_in[24];
    const float* bl2      = (const float*)d_in[25];
    float*       out      = (float*)d_out;

    // workspace layout (floats)
    float* ws = (float*)d_ws;
    size_t off = 0;
    float* xr      = ws + off; off += (size_t)R_REL * N_NODES * HID;  // 16.0M
    float* selfbuf = ws + off; off += (size_t)N_NODES * HID;          //  3.2M
    float* agg     = ws + off; off += (size_t)N_NODES * HID;          //  3.2M
    float* concat  = ws + off; off += (size_t)N_NODES * CAT;          // 12.8M
    float* hidden  = ws + off; off += (size_t)G_PAIRS * MLP_HID;      //  3.2M
    float* Wall    = ws + off; off += (size_t)(R_REL + 1) * HID * HID;

    const int rowTiles = (N_NODES + 15) / 16;
    const int gemmWaves = rowTiles * (HID / 16) * (R_REL + 1);
    const int gemmBlocks = (gemmWaves * 32 + 255) / 256;
    const long long edgeThreads = (long long)N_EDGES * 32;
    const int edgeBlocks = (int)((edgeThreads + 255) / 256);
    const int finBlocks = (N_NODES * HID + 255) / 256;
    const int mlp1Waves = ((G_PAIRS + 15) / 16) * (MLP_HID / 16);
    const int mlp1Blocks = (mlp1Waves * 32 + 255) / 256;
    const int mlp2Blocks = (G_PAIRS * 32 + 255) / 256;

    for (int l = 0; l < 4; ++l) {
        const float* bases = (const float*)d_in[6 + 4 * l];
        const float* comp  = (const float*)d_in[7 + 4 * l];
        const float* loopw = (const float*)d_in[8 + 4 * l];
        const float* bias  = (const float*)d_in[9 + 4 * l];
        const int in_dim   = (l == 0) ? IN_FEAT : HID;
        const float* X     = (l == 0) ? x : (concat + (size_t)(l - 1) * HID);
        const int x_stride = (l == 0) ? IN_FEAT : CAT;

        const int wTotal  = (R_REL + 1) * in_dim * HID;
        build_W<<<(wTotal + 255) / 256, 256, 0, stream>>>(bases, comp, loopw, Wall, in_dim);

        node_gemm_wmma<<<gemmBlocks, 256, 0, stream>>>(X, x_stride, in_dim, Wall, xr, selfbuf);

        hipMemsetAsync(agg, 0, (size_t)N_NODES * HID * sizeof(float), stream);

        edge_scatter<<<edgeBlocks, 256, 0, stream>>>(src, dst, etype, xr, agg);

        finalize_layer<<<finBlocks, 256, 0, stream>>>(agg, selfbuf, bias, concat, l);
    }

    mlp1_wmma<<<mlp1Blocks, 256, 0, stream>>>(concat, user_ids, item_ids, w1, bl1, hidden);
    mlp2_dot<<<mlp2Blocks, 256, 0, stream>>>(hidden, w2, bl2, out);
}